// HeteroGAT_54443005444389
// MI455X (gfx1250) — compile-verified
//
#include <hip/hip_runtime.h>
#include <hip/hip_bf16.h>

#define NN   50000      // nodes
#define NE   800000     // edges per etype
#define NET  3          // edge types
#define NH   4          // heads
#define FIN  256        // input feature dim
#define FHD  64         // per-head dim
#define CDIM 256        // NH*FHD

typedef __attribute__((ext_vector_type(2))) float v2f;
typedef __attribute__((ext_vector_type(8))) float v8f;

// ---------------------------------------------------------------------------
// GEMM via V_WMMA_F32_16X16X4_F32:  Z[N,256] = A[N,K] * B[K,256]
// One wave computes one 16x16 tile of Z.  Lane layout per CDNA5 ISA:
//   A (16x4, 2 VGPRs): lane<16 -> {K=0,K=1}, lane>=16 -> {K=2,K=3}, M = lane&15
//   B (4x16, 2 VGPRs): lane<16 -> rows K=0,1 col=lane; lane>=16 -> rows K=2,3
//   C/D (16x16, 8 VGPRs): vgpr r holds M=r (lanes 0-15) and M=8+r (lanes 16-31)
// ---------------------------------------------------------------------------
__global__ __launch_bounds__(128) void gemm_wmma_f32(
    const float* __restrict__ A, const float* __restrict__ B,
    float* __restrict__ Z, int K)
{
    const int lane = threadIdx.x & 31;
    const int wave = threadIdx.x >> 5;
    const int m0   = blockIdx.x * 16;
    const int n0   = blockIdx.y * 64 + wave * 16;
    const int row  = lane & 15;
    const int hi   = lane >> 4;           // 0 or 1

    v8f acc = {};
    const float* __restrict__ arow = A + (size_t)(m0 + row) * K;

    for (int kk = 0; kk < K; kk += 4) {
        const int k = kk + 2 * hi;
        v2f a, b;
        a.x = arow[k];
        a.y = arow[k + 1];
        b.x = B[(size_t)k       * CDIM + n0 + row];
        b.y = B[(size_t)(k + 1) * CDIM + n0 + row];
        acc = __builtin_amdgcn_wmma_f32_16x16x4_f32(
            /*neg_a=*/false, a, /*neg_b=*/false, b,
            /*c_mod=*/(short)0, acc, /*reuse_a=*/false, /*reuse_b=*/false);
    }

    const int mbase = m0 + hi * 8;
    #pragma unroll
    for (int r = 0; r < 8; ++r)
        Z[(size_t)(mbase + r) * CDIM + n0 + row] = acc[r];
}

// el[n,h] = <z[n,h,:], al[h,:]> ; er likewise
__global__ void attn_scores(const float* __restrict__ Z,
                            const float* __restrict__ al,
                            const float* __restrict__ ar,
                            float* __restrict__ el, float* __restrict__ er)
{
    int idx = blockIdx.x * blockDim.x + threadIdx.x;   // n*NH + h
    if (idx >= NN * NH) return;
    int h = idx & (NH - 1);
    int n = idx >> 2;                                   // NH == 4
    const float* __restrict__ zr = Z + (size_t)n * CDIM + h * FHD;
    const float* __restrict__ alh = al + h * FHD;
    const float* __restrict__ arh = ar + h * FHD;
    float sl = 0.f, sr = 0.f;
    #pragma unroll 8
    for (int f = 0; f < FHD; ++f) {
        float z = zr[f];
        sl += z * alh[f];
        sr += z * arh[f];
    }
    el[idx] = sl;
    er[idx] = sr;
}

__global__ void init_softmax(float* __restrict__ smax, float* __restrict__ denom)
{
    int i = blockIdx.x * blockDim.x + threadIdx.x;
    if (i < NN * NH) {
        smax[i]  = -__builtin_huge_valf();
        denom[i] = 0.f;
    }
}

__global__ void zero_buf(float* __restrict__ p, int n)
{
    int i = blockIdx.x * blockDim.x + threadIdx.x;
    if (i < n) p[i] = 0.f;
}

__device__ __forceinline__ float leaky(float v)
{
    return v > 0.f ? v : 0.2f * v;
}

// sign-aware float atomic max (works for mixed-sign values, init = -inf)
__device__ __forceinline__ void atomicMaxF(float* addr, float v)
{
    if (v >= 0.f) atomicMax((int*)addr, __float_as_int(v));
    else          atomicMin((unsigned int*)addr, __float_as_uint(v));
}

__global__ void edge_max(const int* __restrict__ src, const int* __restrict__ dst,
                         const float* __restrict__ el, const float* __restrict__ er,
                         float* __restrict__ smax)
{
    int e = blockIdx.x * blockDim.x + threadIdx.x;
    if (e >= NE) return;
    int s = src[e], d = dst[e];
    #pragma unroll
    for (int h = 0; h < NH; ++h) {
        float v = leaky(el[s * NH + h] + er[d * NH + h]);
        atomicMaxF(&smax[d * NH + h], v);
    }
}

__global__ void edge_expsum(const int* __restrict__ src, const int* __restrict__ dst,
                            const float* __restrict__ el, const float* __restrict__ er,
                            const float* __restrict__ smax, float* __restrict__ denom)
{
    int e = blockIdx.x * blockDim.x + threadIdx.x;
    if (e >= NE) return;
    int s = src[e], d = dst[e];
    #pragma unroll
    for (int h = 0; h < NH; ++h) {
        float v = leaky(el[s * NH + h] + er[d * NH + h]);
        atomicAdd(&denom[d * NH + h], expf(v - smax[d * NH + h]));
    }
}

// one block per edge; thread c in [0,256) handles channel c = h*64+f
__global__ __launch_bounds__(256) void edge_aggregate(
    const int* __restrict__ src, const int* __restrict__ dst,
    const float* __restrict__ el, const float* __restrict__ er,
    const float* __restrict__ smax, const float* __restrict__ denom,
    const float* __restrict__ Z, float* __restrict__ agg)
{
    int e = blockIdx.x;
    int c = threadIdx.x;
    int h = c >> 6;
    int s = src[e], d = dst[e];
    float v     = leaky(el[s * NH + h] + er[d * NH + h]);
    float alpha = expf(v - smax[d * NH + h]) / denom[d * NH + h];
    atomicAdd(&agg[(size_t)d * CDIM + c], alpha * Z[(size_t)s * CDIM + c]);
}

// out[n,f] = act( mean_h( agg[n,h,f] + sum_e b[e,h,f] ) )
__global__ void head_mean(const float* __restrict__ agg, const float* __restrict__ b,
                          float* __restrict__ hout, int do_relu)
{
    int idx = blockIdx.x * blockDim.x + threadIdx.x;    // n*FHD + f
    if (idx >= NN * FHD) return;
    int f = idx & (FHD - 1);
    int n = idx >> 6;
    float acc = 0.f;
    #pragma unroll
    for (int h = 0; h < NH; ++h) {
        float bs = b[0 * NH * FHD + h * FHD + f]
                 + b[1 * NH * FHD + h * FHD + f]
                 + b[2 * NH * FHD + h * FHD + f];
        acc += agg[(size_t)n * CDIM + h * FHD + f] + bs;
    }
    acc *= (1.f / NH);
    if (do_relu) acc = fmaxf(acc, 0.f);
    hout[idx] = acc;
}

// ---------------------------------------------------------------------------
static void run_layer(const float* hin, int K,
                      const int* src, const int* dst,
                      const float* W, const float* al, const float* ar, const float* b,
                      float* z, float* agg, float* el, float* er,
                      float* smax, float* denom,
                      float* hout, int relu, hipStream_t stream)
{
    const int NHtot = NN * NH;
    zero_buf<<<(NN * CDIM + 255) / 256, 256, 0, stream>>>(agg, NN * CDIM);
    for (int e = 0; e < NET; ++e) {
        gemm_wmma_f32<<<dim3(NN / 16, CDIM / 64), 128, 0, stream>>>(
            hin, W + (size_t)e * K * CDIM, z, K);
        attn_scores<<<(NHtot + 255) / 256, 256, 0, stream>>>(
            z, al + e * NH * FHD, ar + e * NH * FHD, el, er);
        init_softmax<<<(NHtot + 255) / 256, 256, 0, stream>>>(smax, denom);
        edge_max<<<(NE + 255) / 256, 256, 0, stream>>>(
            src + (size_t)e * NE, dst + (size_t)e * NE, el, er, smax);
        edge_expsum<<<(NE + 255) / 256, 256, 0, stream>>>(
            src + (size_t)e * NE, dst + (size_t)e * NE, el, er, smax, denom);
        edge_aggregate<<<NE, 256, 0, stream>>>(
            src + (size_t)e * NE, dst + (size_t)e * NE, el, er, smax, denom, z, agg);
    }
    head_mean<<<(NN * FHD + 255) / 256, 256, 0, stream>>>(agg, b, hout, relu);
}

extern "C" void kernel_launch(void* const* d_in, const int* in_sizes, int n_in,
                              void* d_out, int out_size, void* d_ws, size_t ws_size,
                              hipStream_t stream)
{
    (void)in_sizes; (void)n_in; (void)out_size; (void)ws_size;
    const float* x   = (const float*)d_in[0];
    const int*   src = (const int*)  d_in[1];
    const int*   dst = (const int*)  d_in[2];
    const float* W1  = (const float*)d_in[3];
    const float* al1 = (const float*)d_in[4];
    const float* ar1 = (const float*)d_in[5];
    const float* b1  = (const float*)d_in[6];
    const float* W2  = (const float*)d_in[7];
    const float* al2 = (const float*)d_in[8];
    const float* ar2 = (const float*)d_in[9];
    const float* b2  = (const float*)d_in[10];
    const float* W3  = (const float*)d_in[11];
    const float* al3 = (const float*)d_in[12];
    const float* ar3 = (const float*)d_in[13];
    const float* b3  = (const float*)d_in[14];
    float* out = (float*)d_out;

    // workspace carve-up (floats)
    float* ws    = (float*)d_ws;
    float* z     = ws;                       // NN*CDIM = 12.8M
    float* agg   = z     + (size_t)NN * CDIM; // NN*CDIM
    float* el    = agg   + (size_t)NN * CDIM; // NN*NH
    float* er    = el    + (size_t)NN * NH;
    float* smax  = er    + (size_t)NN * NH;
    float* denom = smax  + (size_t)NN * NH;
    float* hb1   = denom + (size_t)NN * NH;   // NN*FHD
    float* hb2   = hb1   + (size_t)NN * FHD;  // NN*FHD

    run_layer(x,   FIN, src, dst, W1, al1, ar1, b1, z, agg, el, er, smax, denom, hb1, 1, stream);
    run_layer(hb1, FHD, src, dst, W2, al2, ar2, b2, z, agg, el, er, smax, denom, hb2, 1, stream);
    run_layer(hb2, FHD, src, dst, W3, al3, ar3, b3, z, agg, el, er, smax, denom, out, 0, stream);
}